// ContrastiveEmbeddingLoss_13726715478746
// MI455X (gfx1250) — compile-verified
//
#include <hip/hip_runtime.h>

#define N_PTS 8192
#define DIM   128
#define MARGIN 1.0f
#define BLK_ROWS 128
#define BLK_COLS 64
#define LDS_STRIDE 132   // 132 % 64 == 4 -> conflict-free ds_read_b64 for both half-waves

typedef float v2f __attribute__((ext_vector_type(2)));
typedef float v8f __attribute__((ext_vector_type(8)));

#if defined(__has_builtin) && __has_builtin(__builtin_amdgcn_sched_barrier)
#define SCHED_FENCE() __builtin_amdgcn_sched_barrier(0)
#else
#define SCHED_FENCE()
#endif

// ---- pass 1: row squared-norms, and zero the double accumulator ----
__global__ void sq_kernel(const float* __restrict__ x, float* __restrict__ sq,
                          double* __restrict__ wsum) {
    int row = blockIdx.x * blockDim.x + threadIdx.x;
    if (row == 0) *wsum = 0.0;            // stream-ordered before loss_kernel
    if (row < N_PTS) {
        const float4* xr = (const float4*)(x + row * DIM);
        float s = 0.f;
#pragma unroll
        for (int i = 0; i < DIM / 4; ++i) {
            float4 v = xr[i];
            s += v.x * v.x + v.y * v.y + v.z * v.z + v.w * v.w;
        }
        sq[row] = s;
    }
}

// ---- pass 2: fused WMMA gram + contrastive-loss reduction ----
__global__ void __launch_bounds__(256)
loss_kernel(const float* __restrict__ x, const long long* __restrict__ y,
            const float* __restrict__ sq, double* __restrict__ wsum) {
    __shared__ float bs[BLK_COLS * LDS_STRIDE];
    __shared__ float red[256];

    const int tid      = threadIdx.x;
    const int lane     = tid & 31;
    const int wave     = tid >> 5;
    const int rowBlock = blockIdx.y * BLK_ROWS;
    const int colBlock = blockIdx.x * BLK_COLS;

    // Stage B strip (64 cols x 128 K, fp32) into LDS, column-major + pad.
    {
        const float4* xv = (const float4*)x;
#pragma unroll
        for (int i = 0; i < 8; ++i) {
            int linear = tid + i * 256;      // 0..2047 float4 chunks
            int c  = linear >> 5;            // 0..63  (local column)
            int kc = (linear & 31) << 2;     // 0,4,...,124
            float4 v = xv[((colBlock + c) * DIM + kc) >> 2];
            *(float4*)&bs[c * LDS_STRIDE + kc] = v;   // 16B-aligned (528B row pitch)
        }
    }
    __syncthreads();

    const int rowBase = rowBlock + wave * 16;
    const int m   = lane & 15;         // row within A tile / col within B tile
    const int kh  = (lane >> 4) << 1;  // 0 or 2 : k-half per WMMA f32 layout
    const int hi8 = (lane >> 4) << 3;  // 0 or 8 : row offset in C/D layout

    // A strip for this wave: 16 rows x 128 K held as 32 v2f fragments (64 VGPRs).
    v2f a[32];
#pragma unroll
    for (int kk = 0; kk < 32; ++kk)
        a[kk] = *(const v2f*)(x + (rowBase + m) * DIM + kk * 4 + kh);

    // Per-accumulator-row metadata (C/D layout: row = r + hi8).
    float sqi[8];
    int   yi[8];
#pragma unroll
    for (int r = 0; r < 8; ++r) {
        int i = rowBase + r + hi8;
        sqi[r] = sq[i];
        yi[r]  = (int)y[i];
    }

    float lsum = 0.f;
#pragma unroll 1
    for (int ct = 0; ct < 4; ++ct) {
        const int jloc = ct * 16 + m;
        const int j    = colBlock + jloc;
        const float sqj = sq[j];
        const int   yj  = (int)y[j];

        // Batch-load all B fragments for this tile (burst of ds_load_2addr_b64).
        // sched_barrier(0) fences the scheduler so it cannot sink the loads
        // back into the WMMA chain and recycle one destination quad — the
        // whole burst issues first, then a single dscnt wait, then 32 WMMAs
        // back-to-back limited only by the accumulator chain.
        v2f b[32];
        const float* bp = &bs[jloc * LDS_STRIDE + kh];
#pragma unroll
        for (int kk = 0; kk < 32; ++kk)
            b[kk] = *(const v2f*)(bp + kk * 4);

        SCHED_FENCE();

        v8f acc = {};
#pragma unroll
        for (int kk = 0; kk < 32; ++kk)
            acc = __builtin_amdgcn_wmma_f32_16x16x4_f32(
                      false, a[kk], false, b[kk], (short)0, acc, false, false);

        SCHED_FENCE();

        // Fused epilogue: d = ||xi||^2 + ||xj||^2 - 2 xi.xj, hinge for same-class.
#pragma unroll
        for (int r = 0; r < 8; ++r) {
            float d = sqi[r] + sqj - 2.0f * acc[r];
            d = fmaxf(d, 0.0f);
            lsum += (yi[r] == yj) ? fmaxf(MARGIN - d, 0.0f) : d;
        }
    }

    // Block-level tree reduction, then one fp64 atomic per block.
    red[tid] = lsum;
    __syncthreads();
#pragma unroll
    for (int s = 128; s > 0; s >>= 1) {
        if (tid < s) red[tid] += red[tid + s];
        __syncthreads();
    }
    if (tid == 0) atomicAdd(wsum, (double)red[0]);
}

// ---- pass 3: mean ----
__global__ void fin_kernel(const double* __restrict__ wsum, float* __restrict__ out) {
    if (threadIdx.x == 0 && blockIdx.x == 0)
        out[0] = (float)(wsum[0] * (1.0 / ((double)N_PTS * (double)N_PTS)));
}

extern "C" void kernel_launch(void* const* d_in, const int* in_sizes, int n_in,
                              void* d_out, int out_size, void* d_ws, size_t ws_size,
                              hipStream_t stream) {
    const float*     x = (const float*)d_in[0];
    const long long* y = (const long long*)d_in[1];   // reference y is int64
    double* wsum = (double*)d_ws;
    float*  sq   = (float*)((char*)d_ws + 16);        // 16 + 8192*4 bytes of scratch

    sq_kernel<<<N_PTS / 256, 256, 0, stream>>>(x, sq, wsum);

    dim3 grid(N_PTS / BLK_COLS, N_PTS / BLK_ROWS);    // 128 x 64 blocks
    loss_kernel<<<grid, 256, 0, stream>>>(x, y, sq, wsum);

    fin_kernel<<<1, 64, 0, stream>>>(wsum, (float*)d_out);
}